// SoftNCutsLoss_v2_38611755991339
// MI455X (gfx1250) — compile-verified
//
#include <hip/hip_runtime.h>

typedef float v2f __attribute__((ext_vector_type(2)));
typedef float v8f __attribute__((ext_vector_type(8)));

#define F_EPS     2.220446049250313e-16f   // np.finfo(float64).eps
#define INV_SI2   (1.0f / 100.0f)          // 1/sigma_I^2
#define INV_SX2   (1.0f / 16.0f)           // 1/sigma_X^2

__device__ __forceinline__ float shfl32(float v, int srcLane) {
    return __shfl(v, srcLane, 32);
}

__device__ __forceinline__ int clamp31(int v) {
    return min(max(v, 0), 31);
}

// One wave = one 16-voxel strip (b, d, h, w0..w0+15).
// D(16x16) = sum over 45 active (x,y) offset pairs of the banded Toeplitz
// matmul A'(16x24) @ B(24x16) via 6 chained v_wmma_f32_16x16x4_f32.
//   D cols 0..7 = numer(n,k) ; D col 8 = sumw(n)  (B column 8 is all-ones)
// Boundary handling is fully branchless (clamped addresses + cndmask/fma),
// and the B element is formed as bp*bsel+badd so the load cannot be sunk
// into a divergent region — EXEC stays all-ones through every WMMA.
__global__ __launch_bounds__(256) void sncut_stencil(
    const float* __restrict__ batch,   // (4,1,32,32,32)
    const float* __restrict__ preds,   // (4,8,32,32,32)
    float* __restrict__ accA,          // (4,8) workspace
    float* __restrict__ accV)          // (4,8) workspace
{
    const int lane  = threadIdx.x & 31;
    const int wave  = threadIdx.x >> 5;
    const int strip = blockIdx.x * 8 + wave;    // 8192 strips total

    const int w0 = (strip & 1) << 4;
    const int h  = (strip >> 1) & 31;
    const int d  = (strip >> 6) & 31;
    const int b  = strip >> 11;

    const int n   = lane & 15;                  // A-matrix row (voxel in strip)
    const int kk0 = (lane < 16) ? 0 : 2;        // K-slot base within a chunk
    const int col = lane & 15;                  // B-matrix column (class idx)
    const int kc  = col & 7;                    // clamped class for safe loads

    // per-lane B-column constants: bsel kills cols>8, badd makes col 8 = ones
    const float bsel = (col < 8)  ? 1.0f : 0.0f;
    const float badd = (col == 8) ? 1.0f : 0.0f;

    // center intensity for this lane's voxel (used by both half-waves)
    const float Ic = batch[((b * 32 + d) * 32 + h) * 32 + w0 + n];

    v8f acc = {0.0f, 0.0f, 0.0f, 0.0f, 0.0f, 0.0f, 0.0f, 0.0f};

    for (int xi = 0; xi < 7; ++xi) {
        const int   dd  = d + xi - 3;
        const float fdx = (float)(xi - 3);
        for (int yi = 0; yi < 7; ++yi) {
            const int   hh   = h + yi - 3;
            const float fdy  = (float)(yi - 3);
            const float dxy2 = fdx * fdx + fdy * fdy;
            if (dxy2 >= 16.0f) continue;        // whole z-column outside ball (uniform)

            const bool rowIn = ((unsigned)dd < 32u) & ((unsigned)hh < 32u);
            const int  ddc   = clamp31(dd);
            const int  hhc   = clamp31(hh);

            // stage padded batch row: rowb[lane] = batch_pad(dd, hh, w0-3+lane)
            const int   wwr  = w0 - 3 + lane;
            const float rb   = batch[((b * 32 + ddc) * 32 + hhc) * 32 + clamp31(wwr)];
            const float rowb = (rowIn & ((unsigned)wwr < 32u)) ? rb : F_EPS;

            const int prow = ((b * 8 + kc) * 32 + ddc) * 32 + hhc;

            #pragma unroll
            for (int c = 0; c < 6; ++c) {
                const int m0 = c * 4;
                v2f a, bm;
                #pragma unroll
                for (int j = 0; j < 2; ++j) {
                    const int m = m0 + kk0 + j;       // Toeplitz column index, 0..23
                    // ---- A'(n, m) = aff(n, z=m-n): fused dual-gaussian, one exp ----
                    const int   z   = m - n;
                    const float fdz = (float)(z - 3);
                    const float d2  = dxy2 + fdz * fdz;
                    const float Ib  = shfl32(rowb, m);   // batch_pad at (dd,hh,w0+m-3)
                    const float di  = Ic - Ib;
                    const float arg = -(di * di) * INV_SI2 - d2 * INV_SX2;
                    const bool  ok  = ((unsigned)z < 7u) & (d2 < 16.0f);
                    a[j] = ok ? __expf(arg) : 0.0f;
                    // ---- B(m, col) = preds_pad[col, dd, hh, w0+m-3]; col 8 = ones ----
                    const int   ww  = w0 + m - 3;
                    const float bl  = preds[prow * 32 + clamp31(ww)];   // unconditional
                    const float bp  = (rowIn & ((unsigned)ww < 32u)) ? bl : F_EPS;
                    bm[j] = fmaf(bp, bsel, badd);        // consumed by ALL lanes
                }
                // D(16x16) += A_chunk(16x4) x B_chunk(4x16)
                acc = __builtin_amdgcn_wmma_f32_16x16x4_f32(
                    false, a, false, bm, (short)0, acc, false, false);
            }
        }
    }

    // D layout: VGPR v, lane L -> element (M = v + (L>=16 ? 8 : 0), N = L&15).
    // assocA[b,k] += sum_n numer(n,k)*preds(k,n);  assocV[b,k] += sum_n sumw(n)*preds(k,n)
    float pA = 0.0f, pV = 0.0f;
    const int nbase = (lane < 16) ? 0 : 8;
    #pragma unroll
    for (int v = 0; v < 8; ++v) {
        const int   nv = v + nbase;
        const float sw = shfl32(acc[v], (lane & 16) | 8);  // sumw(nv) from column 8
        const float pl = preds[(((b * 8 + kc) * 32 + d) * 32 + h) * 32 + w0 + nv];
        const float p  = pl * bsel;                        // 0 for cols >= 8
        pA += acc[v] * p;
        pV += sw * p;
    }
    pA += __shfl_xor(pA, 16, 32);   // combine the two M-halves (same k)
    pV += __shfl_xor(pV, 16, 32);
    if (lane < 8) {
        atomicAdd(&accA[b * 8 + lane], pA);
        atomicAdd(&accV[b * 8 + lane], pV);
    }
}

__global__ void sncut_finalize(const float* __restrict__ accA,
                               const float* __restrict__ accV,
                               float* __restrict__ out)
{
    const int bb = threadIdx.x;
    if (bb < 4) {
        float s = 0.0f;
        #pragma unroll
        for (int k = 0; k < 8; ++k)
            s += accA[bb * 8 + k] / accV[bb * 8 + k];
        out[bb] = 8.0f - s;
    }
}

extern "C" void kernel_launch(void* const* d_in, const int* in_sizes, int n_in,
                              void* d_out, int out_size, void* d_ws, size_t ws_size,
                              hipStream_t stream)
{
    const float* batch = (const float*)d_in[0];   // (4,1,32,32,32) f32
    const float* preds = (const float*)d_in[1];   // (4,8,32,32,32) f32
    float* accA = (float*)d_ws;                   // 32 floats
    float* accV = accA + 32;                      // 32 floats

    hipMemsetAsync(d_ws, 0, 64 * sizeof(float), stream);
    sncut_stencil<<<1024, 256, 0, stream>>>(batch, preds, accA, accV);
    sncut_finalize<<<1, 32, 0, stream>>>(accA, accV, (float*)d_out);
}